// RealPolarizingBlock_4887672783436
// MI455X (gfx1250) — compile-verified
//
#include <hip/hip_runtime.h>
#include <hip/hip_bf16.h>

#define D_POLAR 1024
#define S_LEN   2048
#define B_N     8
#define KAN_H   32
#define EPSF    1e-8f
#define S_CHUNKS 32
#define S_PER   (S_LEN / S_CHUNKS)       // 64
#define ROWS    (B_N * D_POLAR)          // 8192

typedef float v2f __attribute__((ext_vector_type(2)));
typedef float v8f __attribute__((ext_vector_type(8)));

// ---------------------------------------------------------------------------
// Kernel 1: partial sums over S chunks of x = r*cos(theta), y = r*sin(theta).
// grid = (D/256, S_CHUNKS, B); coalesced across d; deterministic (no atomics).
// ---------------------------------------------------------------------------
__global__ void k_partial(const float* __restrict__ r, const float* __restrict__ th,
                          float* __restrict__ xpart, float* __restrict__ ypart) {
    const int d  = blockIdx.x * blockDim.x + threadIdx.x;
    const int ch = blockIdx.y;
    const int b  = blockIdx.z;
    const int s0 = ch * S_PER;

    float sx = 0.0f, sy = 0.0f;
    const size_t base = ((size_t)b * S_LEN + s0) * D_POLAR + d;
    for (int i = 0; i < S_PER; ++i) {
        const size_t idx = base + (size_t)i * D_POLAR;
        const float rv = r[idx];
        const float tv = th[idx];
        float s, c;
        sincosf(tv, &s, &c);
        sx += rv * c;
        sy += rv * s;
    }
    const int row = b * D_POLAR + d;
    xpart[(size_t)ch * ROWS + row] = sx;
    ypart[(size_t)ch * ROWS + row] = sy;
}

__device__ __forceinline__ float gelu_exact(float v) {
    return 0.5f * v * (1.0f + erff(v * 0.70710678118654752f));
}

// ---------------------------------------------------------------------------
// Kernel 2: finish reduction, run both KAN MLPs. Hidden layers computed per
// lane directly in the V_WMMA_F32_16X16X4_F32 A-matrix layout; second layers
// (K=32 matvecs) run as chains of 8 f32 WMMAs with broadcast-column B.
// One wave handles 16 (b,d) rows. Outputs dx = r_t*cos(theta_t), dy = r_t*sin.
// ---------------------------------------------------------------------------
__global__ void k_mlp(const float* __restrict__ xpart, const float* __restrict__ ypart,
                      const float* __restrict__ mw1, const float* __restrict__ mb1,
                      const float* __restrict__ mw2, const float* __restrict__ mb2,
                      const float* __restrict__ pw1, const float* __restrict__ pb1,
                      const float* __restrict__ pw2, const float* __restrict__ pb2,
                      float* __restrict__ dxw, float* __restrict__ dyw) {
    __shared__ float lm[8][16];
    __shared__ float lp[8][16][2];

    const int lane = threadIdx.x & 31;
    const int w    = threadIdx.x >> 5;
    const int gw   = blockIdx.x * 8 + w;
    const int tileBase = gw * 16;
    const int row  = tileBase + (lane & 15);
    const int off  = (lane < 16) ? 0 : 2;   // K offset per ISA A-matrix 16x4 layout
    const int comp = lane & 1;              // phase output component for this column

    // finish the S reduction for this row (duplicated across half-waves; cheap)
    float sx = 0.0f, sy = 0.0f;
    for (int ch = 0; ch < S_CHUNKS; ++ch) {
        sx += xpart[(size_t)ch * ROWS + row];
        sy += ypart[(size_t)ch * ROWS + row];
    }
    const float invS = 1.0f / (float)S_LEN;
    sx *= invS; sy *= invS;

    const float r_agg  = sqrtf(sx * sx + sy * sy + EPSF);
    const float th_agg = atan2f(sy, sx);
    const float log_r  = logf(r_agg + EPSF);
    float sA, cA;
    sincosf(th_agg, &sA, &cA);

    // hidden activations at the 16 K indices this lane contributes to WMMA A
    float hm[16], hp[16];
#pragma unroll
    for (int c = 0; c < 8; ++c) {
        const int k0 = 4 * c + off;
#pragma unroll
        for (int t = 0; t < 2; ++t) {
            const int k = k0 + t;
            const float vm = log_r * mw1[k] + mb1[k];
            hm[2 * c + t] = gelu_exact(vm);
            const float vp = sA * pw1[k] + cA * pw1[KAN_H + k] + pb1[k];
            hp[2 * c + t] = gelu_exact(vp);
        }
    }

    // second layers: C(16x16) += A(16x4) * B(4x16), chained over K=32.
    // B columns broadcast the weight vector -> every column of C = matvec.
    v8f cm = {};
    v8f cp = {};
#pragma unroll
    for (int kc = 0; kc < 8; ++kc) {
        const int k0 = 4 * kc + off;
        v2f am; am.x = hm[2 * kc]; am.y = hm[2 * kc + 1];
        v2f bm; bm.x = mw2[k0];    bm.y = mw2[k0 + 1];
        cm = __builtin_amdgcn_wmma_f32_16x16x4_f32(false, am, false, bm,
                                                   (short)0, cm, false, false);
        v2f ap; ap.x = hp[2 * kc]; ap.y = hp[2 * kc + 1];
        v2f bp; bp.x = pw2[k0 * 2 + comp]; bp.y = pw2[(k0 + 1) * 2 + comp];
        cp = __builtin_amdgcn_wmma_f32_16x16x4_f32(false, ap, false, bp,
                                                   (short)0, cp, false, false);
    }

    // readout: lane holds column N=(lane&15); rows 0-7 in lanes<16, 8-15 in lanes>=16
    if (lane == 0) {
#pragma unroll
        for (int j = 0; j < 8; ++j) { lm[w][j] = cm[j]; lp[w][j][0] = cp[j]; }
    }
    if (lane == 1) {
#pragma unroll
        for (int j = 0; j < 8; ++j) { lp[w][j][1] = cp[j]; }
    }
    if (lane == 16) {
#pragma unroll
        for (int j = 0; j < 8; ++j) { lm[w][8 + j] = cm[j]; lp[w][8 + j][0] = cp[j]; }
    }
    if (lane == 17) {
#pragma unroll
        for (int j = 0; j < 8; ++j) { lp[w][8 + j][1] = cp[j]; }
    }
    __syncthreads();

    if (lane < 16) {
        const float rt = expf(lm[w][lane] + mb2[0]);
        const float p0 = lp[w][lane][0] + pb2[0];
        const float p1 = lp[w][lane][1] + pb2[1];
        const float nrm = sqrtf(p0 * p0 + p1 * p1 + EPSF);
        const float st = p0 / nrm;
        const float ct = p1 / nrm;
        const float tht = atan2f(st, ct);
        float s2, c2;
        sincosf(tht, &s2, &c2);
        dxw[row] = rt * c2;
        dyw[row] = rt * s2;
    }
}

// ---------------------------------------------------------------------------
// Kernel 3: elementwise output, float4 vectorized.
// out[0 : n)   = r_out
// out[n : 2n)  = theta_out
// ---------------------------------------------------------------------------
__global__ void k_out(const float* __restrict__ r, const float* __restrict__ th,
                      const float* __restrict__ dxw, const float* __restrict__ dyw,
                      float* __restrict__ out) {
    const size_t n  = (size_t)B_N * S_LEN * D_POLAR;
    const size_t i4 = ((size_t)blockIdx.x * blockDim.x + threadIdx.x) * 4;

    const int d = (int)(i4 % D_POLAR);                     // D % 4 == 0, contiguous
    const int b = (int)(i4 / ((size_t)S_LEN * D_POLAR));
    const int rowBase = b * D_POLAR + d;

    const float4 rv  = *(const float4*)(r   + i4);
    const float4 tv  = *(const float4*)(th  + i4);
    const float4 dxv = *(const float4*)(dxw + rowBase);
    const float4 dyv = *(const float4*)(dyw + rowBase);

    float rr[4]  = {rv.x, rv.y, rv.z, rv.w};
    float tt[4]  = {tv.x, tv.y, tv.z, tv.w};
    float dx4[4] = {dxv.x, dxv.y, dxv.z, dxv.w};
    float dy4[4] = {dyv.x, dyv.y, dyv.z, dyv.w};
    float ro[4], to[4];
#pragma unroll
    for (int j = 0; j < 4; ++j) {
        float s, c;
        sincosf(tt[j], &s, &c);
        const float xo = rr[j] * c + dx4[j];
        const float yo = rr[j] * s + dy4[j];
        ro[j] = sqrtf(xo * xo + yo * yo + EPSF);
        to[j] = atan2f(yo, xo);
    }
    *(float4*)(out + i4)     = make_float4(ro[0], ro[1], ro[2], ro[3]);
    *(float4*)(out + n + i4) = make_float4(to[0], to[1], to[2], to[3]);
}

// ---------------------------------------------------------------------------
extern "C" void kernel_launch(void* const* d_in, const int* in_sizes, int n_in,
                              void* d_out, int out_size, void* d_ws, size_t ws_size,
                              hipStream_t stream) {
    const float* r   = (const float*)d_in[0];
    const float* th  = (const float*)d_in[1];
    const float* mw1 = (const float*)d_in[2];
    const float* mb1 = (const float*)d_in[3];
    const float* mw2 = (const float*)d_in[4];
    const float* mb2 = (const float*)d_in[5];
    const float* pw1 = (const float*)d_in[6];
    const float* pb1 = (const float*)d_in[7];
    const float* pw2 = (const float*)d_in[8];
    const float* pb2 = (const float*)d_in[9];
    float* out = (float*)d_out;

    float* ws    = (float*)d_ws;
    float* xpart = ws;                                   // S_CHUNKS * ROWS
    float* ypart = xpart + (size_t)S_CHUNKS * ROWS;      // S_CHUNKS * ROWS
    float* dxw   = ypart + (size_t)S_CHUNKS * ROWS;      // ROWS
    float* dyw   = dxw + ROWS;                           // ROWS

    // 1) partial reduction over S
    dim3 g1(D_POLAR / 256, S_CHUNKS, B_N);
    k_partial<<<g1, 256, 0, stream>>>(r, th, xpart, ypart);

    // 2) aggregate + KAN MLPs (WMMA second layers): 8192 rows / (16 rows * 8 waves)
    k_mlp<<<ROWS / 128, 256, 0, stream>>>(xpart, ypart, mw1, mb1, mw2, mb2,
                                          pw1, pb1, pw2, pb2, dxw, dyw);

    // 3) elementwise output, float4 vectorized
    const size_t n = (size_t)B_N * S_LEN * D_POLAR;
    k_out<<<(unsigned)(n / 4 / 256), 256, 0, stream>>>(r, th, dxw, dyw, out);
}